// GNNClassifier_88648124990386
// MI455X (gfx1250) — compile-verified
//
#include <hip/hip_runtime.h>

#define N_NODES   100000
#define N_EDGES   1200000
#define EMBED_D   32
#define HIDDEN_D  64
#define N_CLASSES 10
#define N_GRAPHS  1024

typedef __attribute__((ext_vector_type(2))) float v2f;
typedef __attribute__((ext_vector_type(8))) float v8f;

// Native L2-side fp32 atomic add, device scope (coherent across WGPs).
// Non-returning -> STOREcnt; s_endpgm's implicit wait-idle covers completion.
__device__ __forceinline__ void atomAdd1(float* p, float v) {
    asm volatile("global_atomic_add_f32 %0, %1, off scope:SCOPE_DEV"
                 :: "v"(p), "v"(v) : "memory");
}
__device__ __forceinline__ void atomAdd4(float* p, float4 v) {
    asm volatile(
        "global_atomic_add_f32 %0, %1, off scope:SCOPE_DEV\n\t"
        "global_atomic_add_f32 %0, %2, off offset:4 scope:SCOPE_DEV\n\t"
        "global_atomic_add_f32 %0, %3, off offset:8 scope:SCOPE_DEV\n\t"
        "global_atomic_add_f32 %0, %4, off offset:12 scope:SCOPE_DEV"
        :: "v"(p), "v"(v.x), "v"(v.y), "v"(v.z), "v"(v.w) : "memory");
}

// ---------------------------------------------------------------- zero fill
__global__ void zero_kernel(float* __restrict__ p, int n) {
    int t = blockIdx.x * blockDim.x + threadIdx.x;
    if (t < n) p[t] = 0.0f;
}

// ------------------------------------------------------- embedding gather
__global__ void gather_embed_kernel(const int* __restrict__ ids,
                                    const float* __restrict__ embed,
                                    float* __restrict__ x0, int nNodes) {
    int t = blockIdx.x * blockDim.x + threadIdx.x;
    if (t >= nNodes * (EMBED_D / 4)) return;
    int node = t >> 3;          // EMBED_D/4 == 8
    int q    = t & 7;
    float4 v = reinterpret_cast<const float4*>(embed + (size_t)ids[node] * EMBED_D)[q];
    reinterpret_cast<float4*>(x0 + (size_t)node * EMBED_D)[q] = v;
}

// ------------------------------------------------------------ edge degree
__global__ void edge_count_kernel(const int* __restrict__ dst,
                                  float* __restrict__ cnt, int nEdges) {
    int e = blockIdx.x * blockDim.x + threadIdx.x;
    if (e >= nEdges) return;
    atomAddF_guard:
    atomAdd1(cnt + dst[e], 1.0f);
}

// -------------------------------------------------------- edge scatter-add
// agg[dst[e], :] += x[src[e], :]  ; one thread per (edge, 4 dims)
template <int DIM>
__global__ void edge_scatter_kernel(const int* __restrict__ src,
                                    const int* __restrict__ dst,
                                    const float* __restrict__ x,
                                    float* __restrict__ agg, int nEdges) {
    constexpr int QPE = DIM / 4;
    int t = blockIdx.x * blockDim.x + threadIdx.x;
    if (t >= nEdges * QPE) return;
    int e = t / QPE;            // QPE is a power of two -> shifts
    int q = t & (QPE - 1);
    int s = src[e], d = dst[e];
    float4 v = reinterpret_cast<const float4*>(x + (size_t)s * DIM)[q];
    atomAdd4(agg + (size_t)d * DIM + q * 4, v);
}

// ------------------------------------------------------------- SAGE GEMM
// out = relu( (aggRaw / max(cnt,1)) @ Wl + bias + x @ Wr )
// 128 threads = 4 waves; wave w -> column tile w (16 cols); block -> 16 rows.
// V_WMMA_F32_16X16X4_F32 fragment layout (ISA 7.12.2):
//   A 16x4 : lane l(0..15) row M=l ; VGPR j, lane-group g -> K = k0 + j + 2g
//   B 4x16 : lane l(0..15) col N=l ; VGPR j, lane-group g -> K = k0 + j + 2g
//   C 16x16: VGPR v, lane-group g  -> M = v + 8g ; N = lane&15
template <int K>
__global__ void sage_gemm_kernel(const float* __restrict__ aggRaw,
                                 const float* __restrict__ cnt,
                                 const float* __restrict__ x,
                                 const float* __restrict__ Wl,
                                 const float* __restrict__ Wr,
                                 const float* __restrict__ bias,
                                 float* __restrict__ out, int nNodes) {
    const int lane = threadIdx.x & 31;
    const int wave = threadIdx.x >> 5;   // 0..3 : column tile
    const int row0 = blockIdx.x * 16;
    if (row0 >= nNodes) return;          // block-uniform; EXEC stays all-ones
    const int col0 = wave * 16;
    const int grp  = lane >> 4;          // 0 or 1
    const int lr   = lane & 15;

    const int   arow = row0 + lr;
    const float invc = 1.0f / fmaxf(cnt[arow], 1.0f);

    v8f c = {};
    // mean(neighbors) @ Wl  (mean folded into A load: row scaling commutes)
#pragma unroll
    for (int k0 = 0; k0 < K; k0 += 4) {
        const int ka = k0 + 2 * grp;
        v2f a, b;
        a.x = aggRaw[(size_t)arow * K + ka + 0] * invc;
        a.y = aggRaw[(size_t)arow * K + ka + 1] * invc;
        b.x = Wl[(size_t)(ka + 0) * HIDDEN_D + col0 + lr];
        b.y = Wl[(size_t)(ka + 1) * HIDDEN_D + col0 + lr];
        c = __builtin_amdgcn_wmma_f32_16x16x4_f32(false, a, false, b,
                                                  (short)0, c, false, false);
    }
    // x @ Wr
#pragma unroll
    for (int k0 = 0; k0 < K; k0 += 4) {
        const int ka = k0 + 2 * grp;
        v2f a, b;
        a.x = x[(size_t)arow * K + ka + 0];
        a.y = x[(size_t)arow * K + ka + 1];
        b.x = Wr[(size_t)(ka + 0) * HIDDEN_D + col0 + lr];
        b.y = Wr[(size_t)(ka + 1) * HIDDEN_D + col0 + lr];
        c = __builtin_amdgcn_wmma_f32_16x16x4_f32(false, a, false, b,
                                                  (short)0, c, false, false);
    }

    const float bv = bias[col0 + lr];
#pragma unroll
    for (int v = 0; v < 8; ++v) {
        const int m = v + 8 * grp;
        out[(size_t)(row0 + m) * HIDDEN_D + col0 + lr] = fmaxf(c[v] + bv, 0.0f);
    }
}

// ----------------------------------------------------------- graph pooling
__global__ void pool_kernel(const int* __restrict__ batch,
                            const float* __restrict__ h,
                            float* __restrict__ gsum,
                            float* __restrict__ gcnt, int nNodes) {
    int t = blockIdx.x * blockDim.x + threadIdx.x;
    if (t >= nNodes * (HIDDEN_D / 4)) return;
    int node = t >> 4;                  // HIDDEN_D/4 == 16
    int q    = t & 15;
    int g    = batch[node];
    float4 v = reinterpret_cast<const float4*>(h + (size_t)node * HIDDEN_D)[q];
    atomAdd4(gsum + (size_t)g * HIDDEN_D + q * 4, v);
    if (q == 0) atomAdd1(gcnt + g, 1.0f);
}

// ------------------------------------------------------------ output head
__global__ void head_kernel(const float* __restrict__ gsum,
                            const float* __restrict__ gcnt,
                            const float* __restrict__ Wo,
                            const float* __restrict__ bo,
                            float* __restrict__ out) {
    int t = blockIdx.x * blockDim.x + threadIdx.x;
    if (t >= N_GRAPHS * N_CLASSES) return;
    int g = t / N_CLASSES;
    int c = t - g * N_CLASSES;
    float invc = 1.0f / fmaxf(gcnt[g], 1.0f);
    float acc = 0.0f;
    for (int d = 0; d < HIDDEN_D; ++d)
        acc += gsum[(size_t)g * HIDDEN_D + d] * invc * Wo[(size_t)d * N_CLASSES + c];
    out[t] = acc + bo[c];
}

// ----------------------------------------------------------------- launch
extern "C" void kernel_launch(void* const* d_in, const int* in_sizes, int n_in,
                              void* d_out, int out_size, void* d_ws, size_t ws_size,
                              hipStream_t stream) {
    const int*   node_ids = (const int*)  d_in[0];
    const int*   src      = (const int*)  d_in[1];
    const int*   dst      = (const int*)  d_in[2];
    const int*   batch    = (const int*)  d_in[3];
    const float* embed    = (const float*)d_in[4];
    const float* W1_l     = (const float*)d_in[5];
    const float* b1       = (const float*)d_in[6];
    const float* W1_r     = (const float*)d_in[7];
    const float* W2_l     = (const float*)d_in[8];
    const float* b2       = (const float*)d_in[9];
    const float* W2_r     = (const float*)d_in[10];
    const float* W_out    = (const float*)d_in[11];
    const float* b_out    = (const float*)d_in[12];
    float* out = (float*)d_out;

    // workspace layout (floats)
    float* ws   = (float*)d_ws;
    float* x0   = ws;                                 // N*32  fully written
    float* h1   = x0   + (size_t)N_NODES * EMBED_D;   // N*64
    float* h2   = h1   + (size_t)N_NODES * HIDDEN_D;  // N*64
    float* agg1 = h2   + (size_t)N_NODES * HIDDEN_D;  // N*32  zeroed
    float* agg2 = agg1 + (size_t)N_NODES * EMBED_D;   // N*64  zeroed
    float* cnt  = agg2 + (size_t)N_NODES * HIDDEN_D;  // N     zeroed
    float* gsum = cnt  + (size_t)N_NODES;             // G*64  zeroed
    float* gcnt = gsum + (size_t)N_GRAPHS * HIDDEN_D; // G     zeroed
    const int zeroN = N_NODES * EMBED_D + N_NODES * HIDDEN_D + N_NODES
                    + N_GRAPHS * HIDDEN_D + N_GRAPHS;

    const int B = 256;
    auto blocks = [](int n, int b) { return (n + b - 1) / b; };

    zero_kernel<<<blocks(zeroN, B), B, 0, stream>>>(agg1, zeroN);

    gather_embed_kernel<<<blocks(N_NODES * 8, B), B, 0, stream>>>(
        node_ids, embed, x0, N_NODES);

    edge_count_kernel<<<blocks(N_EDGES, B), B, 0, stream>>>(dst, cnt, N_EDGES);

    // layer 1: aggregate x0 (d=32), fused SAGE GEMM + relu -> h1
    edge_scatter_kernel<EMBED_D><<<blocks(N_EDGES * (EMBED_D / 4), B), B, 0, stream>>>(
        src, dst, x0, agg1, N_EDGES);
    sage_gemm_kernel<EMBED_D><<<N_NODES / 16, 128, 0, stream>>>(
        agg1, cnt, x0, W1_l, W1_r, b1, h1, N_NODES);

    // layer 2: aggregate h1 (d=64), fused SAGE GEMM + relu -> h2
    edge_scatter_kernel<HIDDEN_D><<<blocks(N_EDGES * (HIDDEN_D / 4), B), B, 0, stream>>>(
        src, dst, h1, agg2, N_EDGES);
    sage_gemm_kernel<HIDDEN_D><<<N_NODES / 16, 128, 0, stream>>>(
        agg2, cnt, h1, W2_l, W2_r, b2, h2, N_NODES);

    // global mean pool + linear head
    pool_kernel<<<blocks(N_NODES * (HIDDEN_D / 4), B), B, 0, stream>>>(
        batch, h2, gsum, gcnt, N_NODES);
    head_kernel<<<blocks(N_GRAPHS * N_CLASSES, B), B, 0, stream>>>(
        gsum, gcnt, W_out, b_out, out);
}